// LJ_81123342287652
// MI455X (gfx1250) — compile-verified
//
#include <hip/hip_runtime.h>

typedef __attribute__((ext_vector_type(2))) float v2f;
typedef __attribute__((ext_vector_type(4))) float v4f;
typedef __attribute__((ext_vector_type(8))) float v8f;

#define NPART   512
#define NTILE   32          // 512 / 16 column-groups
#define THREADS 256         // 8 wave32 per workgroup
#define SPLIT   4           // blocks per batch; each block does 8 tile-rows

// ---------------------------------------------------------------------------
// Kernel 1: pairwise LJ sums. grid = B*SPLIT blocks. Each wave owns one
// tile-row (ti) and sweeps all 32 tile-cols.
// D = (-2A)*B + (qm+qn[+inf on diag]) = d^2 straight out of the WMMA;
// rcp(+inf)=0 makes diagonal pairs contribute exactly zero -> no masking.
// Unroll is capped at 4 so the scheduler cannot keep 32 accumulator sets
// live (round 3 pushed VGPR usage past 256 and wrecked occupancy).
// ---------------------------------------------------------------------------
__global__ __launch_bounds__(THREADS)
void lj_pair_kernel(const float* __restrict__ x, float* __restrict__ partial) {
    // Operand image: for column-group t, lane l (particle p = t*16 + (l&15)):
    //   l<16 : {x[p], y[p], q[p], 0}   (WMMA B rows K=0,K=1)
    //   l>=16: {z[p], 0,    q[p], 0}   (WMMA B rows K=2,K=3 pad)
    __shared__ v4f  op[NTILE * 32];     // 16 KB
    __shared__ float red[THREADS];

    const int blk  = blockIdx.x;
    const int b    = blk / SPLIT;
    const int part = blk % SPLIT;
    const int tid  = threadIdx.x;
    const float* __restrict__ xb = x + (size_t)b * NPART * 3;

    // ---- Build the operand image (each thread fills 4 entries)
    #pragma unroll
    for (int i = tid; i < NTILE * 32; i += THREADS) {
        const int t = i >> 5;
        const int l = i & 31;
        const int p = t * 16 + (l & 15);
        float px = xb[p * 3 + 0];
        float py = xb[p * 3 + 1];
        float pz = xb[p * 3 + 2];
        float q  = px * px + py * py + pz * pz;
        v4f v;
        if (l < 16) { v.x = px; v.y = py; }
        else        { v.x = pz; v.y = 0.0f; }
        v.z = q; v.w = 0.0f;
        op[i] = v;
    }
    __syncthreads();

    const int  lane      = tid & 31;
    const int  wave      = tid >> 5;
    const int  l16       = lane & 15;
    const int  mbase_loc = (lane >= 16) ? 8 : 0;
    const int  ti        = part * 8 + wave;                    // wave's tile-row
    const int  tiu       = __builtin_amdgcn_readfirstlane(ti); // scalar copy

    // A operand for tile-row ti, pre-scaled by -2 (hoisted)
    v4f av = op[ti * 32 + lane];
    v2f a; a.x = av.x * -2.0f; a.y = av.y * -2.0f;

    // q for the 8 M-rows this lane's accumulator VGPRs map to (pairs, hoisted)
    v2f qm2[4];
    #pragma unroll
    for (int k = 0; k < 4; ++k) {
        qm2[k].x = op[ti * 32 + mbase_loc + 2 * k + 0].z;
        qm2[k].y = op[ti * 32 + mbase_loc + 2 * k + 1].z;
    }

    // +inf bias at this lane's (at most one) diagonal slot; 0 elsewhere.
    const int dr = l16 - mbase_loc;     // diagonal r for this lane (may be OOR)
    v2f dbias[4];
    #pragma unroll
    for (int k = 0; k < 4; ++k) {
        dbias[k].x = (2 * k + 0 == dr) ? __builtin_inff() : 0.0f;
        dbias[k].y = (2 * k + 1 == dr) ? __builtin_inff() : 0.0f;
    }

    v2f s6a  = {0.f, 0.f}, s6b  = {0.f, 0.f};   // sums of r^-6
    v2f s12a = {0.f, 0.f}, s12b = {0.f, 0.f};   // sums of r^-12

    #pragma clang loop unroll_count(4)
    for (int tj = 0; tj < NTILE; ++tj) {
        v4f bv = op[tj * 32 + lane];            // ds_load_b96
        v2f bb; bb.x = bv.x; bb.y = bv.y;
        v2f qn2; qn2.x = bv.z; qn2.y = bv.z;

        // C = qm + qn  (broadcast pk_adds)
        v2f c0 = qm2[0] + qn2, c1 = qm2[1] + qn2;
        v2f c2 = qm2[2] + qn2, c3 = qm2[3] + qn2;
        if (tj == tiu) {                        // uniform; 1-in-32 tiles
            c0 = c0 + dbias[0]; c1 = c1 + dbias[1];
            c2 = c2 + dbias[2]; c3 = c3 + dbias[3];
        }
        v8f c = {c0.x, c0.y, c1.x, c1.y, c2.x, c2.y, c3.x, c3.y};

        // d^2 (diag slots = +inf)
        v8f d2 = __builtin_amdgcn_wmma_f32_16x16x4_f32(
                     false, a, false, bb, (short)0, c, false, false);

        #pragma unroll
        for (int r = 0; r < 8; r += 2) {
            v2f dd;  dd.x = d2[r]; dd.y = d2[r + 1];
            v2f inv2;                                    // rcp(+inf) = +0
            inv2.x = __builtin_amdgcn_rcpf(dd.x);
            inv2.y = __builtin_amdgcn_rcpf(dd.y);
            v2f p6 = (inv2 * inv2) * inv2;               // r^-6 (0 on diag)
            if ((r & 2) == 0) { s12a = p6 * p6 + s12a; s6a = s6a + p6; }
            else              { s12b = p6 * p6 + s12b; s6b = s6b + p6; }
        }
    }

    // e = 4 * (sum r^-12 - sum r^-6), ordered pairs
    v2f s12 = s12a + s12b;
    v2f s6  = s6a + s6b;
    float e = 4.0f * ((s12.x + s12.y) - (s6.x + s6.y));

    // ---- deterministic tree reduction over the block
    red[tid] = e;
    __syncthreads();
    #pragma unroll
    for (int s = THREADS / 2; s > 0; s >>= 1) {
        if (tid < s) red[tid] += red[tid + s];
        __syncthreads();
    }
    if (tid == 0) partial[blk] = red[0];
}

// ---------------------------------------------------------------------------
// Kernel 2: per-batch finalize. Combines SPLIT partials (fixed order ->
// deterministic) and adds the harmonic COM restraint computed from x.
// ---------------------------------------------------------------------------
__global__ __launch_bounds__(THREADS)
void lj_finalize_kernel(const float* __restrict__ x,
                        const float* __restrict__ partial,
                        float* __restrict__ out) {
    __shared__ float red[THREADS];
    const int b   = blockIdx.x;
    const int tid = threadIdx.x;
    const float* __restrict__ xb = x + (size_t)b * NPART * 3;

    float sx = 0.f, sy = 0.f, sz = 0.f, qq = 0.f;
    #pragma unroll
    for (int p = tid; p < NPART; p += THREADS) {
        float px = xb[p * 3 + 0];
        float py = xb[p * 3 + 1];
        float pz = xb[p * 3 + 2];
        sx += px; sy += py; sz += pz;
        qq += px * px + py * py + pz * pz;
    }

    float vals[4] = {sx, sy, sz, qq};
    float tot[4];
    #pragma unroll
    for (int k = 0; k < 4; ++k) {
        __syncthreads();
        red[tid] = vals[k];
        __syncthreads();
        for (int s = THREADS / 2; s > 0; s >>= 1) {
            if (tid < s) red[tid] += red[tid + s];
            __syncthreads();
        }
        tot[k] = red[0];
    }

    if (tid == 0) {
        float pair = (partial[b * SPLIT + 0] + partial[b * SPLIT + 1]) +
                     (partial[b * SPLIT + 2] + partial[b * SPLIT + 3]);
        float s2   = tot[0] * tot[0] + tot[1] * tot[1] + tot[2] * tot[2];
        // 0.5*K*sum||x-com||^2 = 0.25*(Q - ||S||^2/N), K=0.5
        float harm = 0.25f * (tot[3] - s2 * (1.0f / (float)NPART));
        out[b] = -(0.5f * pair + harm);
    }
}

extern "C" void kernel_launch(void* const* d_in, const int* in_sizes, int n_in,
                              void* d_out, int out_size, void* d_ws, size_t ws_size,
                              hipStream_t stream) {
    const float* x = (const float*)d_in[0];
    float* out     = (float*)d_out;
    float* partial = (float*)d_ws;             // B*SPLIT floats of scratch
    const int B = in_sizes[0] / (NPART * 3);   // 128

    lj_pair_kernel<<<B * SPLIT, THREADS, 0, stream>>>(x, partial);
    lj_finalize_kernel<<<B, THREADS, 0, stream>>>(x, partial, out);
}